// DKNN_24988119728299
// MI455X (gfx1250) — compile-verified
//
#include <hip/hip_runtime.h>

typedef float v2f __attribute__((ext_vector_type(2)));
typedef float v8f __attribute__((ext_vector_type(8)));

// global -> LDS async copies (GLOBAL_LOAD_ASYNC_TO_LDS_*, tracked by ASYNCcnt)
__device__ __forceinline__ void async_copy_b128(unsigned lds_off, const void* gptr)
{
    asm volatile("global_load_async_to_lds_b128 %0, %1, off"
                 :: "v"(lds_off), "v"((unsigned long long)(uintptr_t)gptr)
                 : "memory");
}
__device__ __forceinline__ void async_copy_b32(unsigned lds_off, const void* gptr)
{
    asm volatile("global_load_async_to_lds_b32 %0, %1, off"
                 :: "v"(lds_off), "v"((unsigned long long)(uintptr_t)gptr)
                 : "memory");
}
__device__ __forceinline__ void async_wait0()
{
    asm volatile("s_wait_asynccnt 0" ::: "memory");
}
// generic __shared__ pointer -> raw LDS byte offset (flat LDS aperture keeps it in addr[31:0])
__device__ __forceinline__ unsigned lds_offset(const void* p)
{
    return (unsigned)(uintptr_t)p;
}

// ---------------------------------------------------------------------------
// Generic fp32 WMMA GEMM: out = epilogue(A[MxK] @ B + ...)
//   MODE 0: B=[KxN] row-major weights; out = PReLU(A@B + bias, alpha)
//   MODE 1: B=[KxN];                   out = A@B + bias
//   MODE 2: B=[KxN];                   out = A@B + A          (K==N residual)
//   MODE 3: B=[NxK] (A@B^T);           out = (A@B^T) * scale          (pe_sims)
//   MODE 4: B=[NxK] (A@B^T);           pe = out[row][col] (pre-stored pe_sims);
//                                      out = (pe < thresh[row]) ? 0 : (A@B^T)*scale
// Block: 256 threads (8 waves). Block tile 64x128; each wave owns 32x32
// (4 x 16x16 f32 WMMA accumulators). Double-buffered LDS; staging is done
// entirely with ASYNCcnt-tracked global->LDS copies issued one K-chunk ahead
// so the copy of chunk i+1 overlaps the WMMA stream of chunk i.
// ---------------------------------------------------------------------------
template <int MODE>
__global__ __launch_bounds__(256) void gemm_wmma(
    const float* __restrict__ A, const float* __restrict__ B,
    const float* __restrict__ bias, float* out,
    int M, int N, int K, const float* __restrict__ alphap, float scale,
    const float* __restrict__ thresh)
{
    constexpr int BM = 64, BN = 128, BK = 32, LDSS = BK + 4; // stride 36: 16B-aligned rows
    __shared__ float As[2][BM][LDSS];
    __shared__ float Bs[2][BN][LDSS];

    const int tid  = threadIdx.x;
    const int wave = tid >> 5;
    const int lane = tid & 31;
    const int wm   = wave & 1;    // 0..1 -> 32-row sub-tile
    const int wn   = wave >> 1;   // 0..3 -> 32-col sub-tile
    const int l16  = lane & 15;
    const int lh   = lane >> 4;   // 0/1

    const int m0 = blockIdx.y * BM;
    const int n0 = blockIdx.x * BN;

    v8f acc[2][2] = {};

    // ---- async staging of one K-chunk into LDS buffer `buf` ----
    auto stage = [&](int buf, int k0) {
        {   // A tile: 64 x 32, 16B per lane, 2 per thread
            const int r = tid >> 3, c = (tid & 7) << 2;
            #pragma unroll
            for (int rr = 0; rr < BM; rr += 32)
                async_copy_b128(lds_offset(&As[buf][r + rr][c]),
                                A + (size_t)(m0 + r + rr) * K + k0 + c);
        }
        if (MODE >= 3) {
            // B is [N x K] row-major (A@B^T): contiguous rows -> 16B async copies
            const int r = tid >> 3, c = (tid & 7) << 2;
            #pragma unroll
            for (int rr = 0; rr < BN; rr += 32)
                async_copy_b128(lds_offset(&Bs[buf][r + rr][c]),
                                B + (size_t)(n0 + r + rr) * K + k0 + c);
        } else {
            // B is [K x N] weights: transpose into Bs[n][k] via b32 async scatter
            const int kq = tid >> 6;             // 0..3
            const int nn = (tid & 63) << 1;      // 0,2,..,126
            #pragma unroll
            for (int kk2 = 0; kk2 < 8; ++kk2) {
                const int k = kq + (kk2 << 2);   // 0..31
                const float* g = B + (size_t)(k0 + k) * N + n0 + nn;
                async_copy_b32(lds_offset(&Bs[buf][nn + 0][k]), g + 0);
                async_copy_b32(lds_offset(&Bs[buf][nn + 1][k]), g + 1);
            }
        }
    };

    // prologue: fill buffer 0
    stage(0, 0);
    async_wait0();
    __syncthreads();

    const int nk = K / BK;
    int buf = 0;
    for (int i = 0; i < nk; ++i) {
        // issue next chunk's copies; they run while we do WMMA on this chunk
        if (i + 1 < nk) stage(buf ^ 1, (i + 1) * BK);

        // ---- 8 K-steps x 4 WMMA (V_WMMA_F32_16X16X4_F32) ----
        #pragma unroll
        for (int kk = 0; kk < BK; kk += 4) {
            const int kf = kk + (lh << 1);
            v2f a[2], b[2];
            a[0] = *(const v2f*)&As[buf][wm * 32 + l16][kf];
            a[1] = *(const v2f*)&As[buf][wm * 32 + 16 + l16][kf];
            b[0] = *(const v2f*)&Bs[buf][wn * 32 + l16][kf];
            b[1] = *(const v2f*)&Bs[buf][wn * 32 + 16 + l16][kf];
            #pragma unroll
            for (int mi = 0; mi < 2; ++mi)
                #pragma unroll
                for (int ni = 0; ni < 2; ++ni)
                    acc[mi][ni] = __builtin_amdgcn_wmma_f32_16x16x4_f32(
                        false, a[mi], false, b[ni], (short)0, acc[mi][ni],
                        false, false);
        }

        async_wait0();     // own copies into next buffer done
        __syncthreads();   // everyone's copies visible, reads of this buffer done
        buf ^= 1;
    }

    // ---- epilogue: C reg v, lane l -> row = v + 8*lh, col = l16 ----
    const float alpha = (MODE == 0) ? alphap[0] : 0.0f;
    #pragma unroll
    for (int mi = 0; mi < 2; ++mi) {
        #pragma unroll
        for (int ni = 0; ni < 2; ++ni) {
            #pragma unroll
            for (int v = 0; v < 8; ++v) {
                const int row = m0 + wm * 32 + mi * 16 + v + lh * 8;
                const int col = n0 + wn * 32 + ni * 16 + l16;
                float x = acc[mi][ni][v];
                if (MODE == 0 || MODE == 1) {
                    x += bias[col];
                    if (MODE == 0) x = (x >= 0.0f) ? x : alpha * x;
                } else if (MODE == 2) {
                    x += A[(size_t)row * K + col];
                } else if (MODE == 3) {
                    x *= scale;
                } else { // MODE 4
                    const float th = thresh[row];
                    const float pe = out[(size_t)row * N + col];
                    x = (pe < th) ? 0.0f : x * scale;
                }
                out[(size_t)row * N + col] = x;
            }
        }
    }
}

// ---------------------------------------------------------------------------
// x[i] = 0.5*x[i] + 0.5*pe[i]   (q_in / k_in mix, PE_WEIGHT = 0.5)
// ---------------------------------------------------------------------------
__global__ void mix_kernel(float* x, const float* __restrict__ pe, int n)
{
    const int i = blockIdx.x * blockDim.x + threadIdx.x;
    if (i < n) x[i] = 0.5f * (x[i] + pe[i]);
}

// ---------------------------------------------------------------------------
// Per-row exact k-th largest via 32-bit radix bisection on order-preserving
// uint keys. One block (256 thr) per row; row (4096 f32) held in 16KB LDS.
// ---------------------------------------------------------------------------
__global__ __launch_bounds__(256) void select_kernel(
    const float* __restrict__ sims, float* __restrict__ thresh, int N, int k)
{
    __shared__ unsigned keys[4096];
    __shared__ int red[256];
    const int row = blockIdx.x;
    const int tid = threadIdx.x;

    for (int i = tid; i < N; i += 256) {
        const unsigned u = __float_as_uint(sims[(size_t)row * N + i]);
        keys[i] = (u & 0x80000000u) ? ~u : (u | 0x80000000u);
    }
    __syncthreads();

    unsigned prefix = 0;
    int want = k;
    for (int bit = 31; bit >= 0; --bit) {
        const unsigned bm = 1u << bit;
        const unsigned hi = ~((bm << 1) - 1u);  // bits above `bit` (0 for bit 31)
        int c = 0;
        for (int i = tid; i < N; i += 256) {
            const unsigned kk = keys[i];
            if (((kk & hi) == prefix) && (kk & bm)) ++c;
        }
        red[tid] = c;
        __syncthreads();
        for (int s = 128; s > 0; s >>= 1) {
            if (tid < s) red[tid] += red[tid + s];
            __syncthreads();
        }
        const int cnt = red[0];
        __syncthreads();
        if (cnt >= want) prefix |= bm; else want -= cnt;
    }

    if (tid == 0) {
        const unsigned u =
            (prefix & 0x80000000u) ? (prefix & 0x7FFFFFFFu) : ~prefix;
        thresh[row] = __uint_as_float(u);
    }
}

// ---------------------------------------------------------------------------
extern "C" void kernel_launch(void* const* d_in, const int* in_sizes, int n_in,
                              void* d_out, int out_size, void* d_ws, size_t ws_size,
                              hipStream_t stream)
{
    const float* feat_q  = (const float*)d_in[0];   // [8192,128]
    const float* feat_kv = (const float*)d_in[1];   // [4096,128]
    const float* pe_q    = (const float*)d_in[2];   // [8192,256]
    const float* pe_kv   = (const float*)d_in[3];   // [4096,256]
    const float* W1 = (const float*)d_in[4];
    const float* b1 = (const float*)d_in[5];
    const float* W2 = (const float*)d_in[6];
    const float* b2 = (const float*)d_in[7];
    const float* W3 = (const float*)d_in[8];
    const float* b3 = (const float*)d_in[9];
    const float* alphap = (const float*)d_in[10];   // scalar on device
    const float* Wq = (const float*)d_in[11];
    const float* Wk = (const float*)d_in[12];

    float* out = (float*)d_out;                     // [8192,4096]

    const int BQ = 8192, NK = 4096, R = BQ + NK;    // 12288 rows
    const int DM = 256, DI = 128;
    const float scale = 1.0f / 16.0f;               // 1/sqrt(256)

    float* bufA = (float*)d_ws;                     // [12288,256]
    float* bufB = bufA + (size_t)R * DM;            // [12288,256]
    float* thr  = bufB + (size_t)R * DM;            // [8192]

    const dim3 blk(256);

    // MLP layer 1: bufA = PReLU(feat @ W1 + b1)
    gemm_wmma<0><<<dim3(DM / 128, BQ / 64), blk, 0, stream>>>(
        feat_q, W1, b1, bufA, BQ, DM, DI, alphap, scale, nullptr);
    gemm_wmma<0><<<dim3(DM / 128, NK / 64), blk, 0, stream>>>(
        feat_kv, W1, b1, bufA + (size_t)BQ * DM, NK, DM, DI, alphap, scale, nullptr);

    // MLP layer 2: bufB = PReLU(bufA @ W2 + b2)
    gemm_wmma<0><<<dim3(DM / 128, R / 64), blk, 0, stream>>>(
        bufA, W2, b2, bufB, R, DM, DM, alphap, scale, nullptr);

    // MLP layer 3: bufA = bufB @ W3 + b3   (== ae)
    gemm_wmma<1><<<dim3(DM / 128, R / 64), blk, 0, stream>>>(
        bufB, W3, b3, bufA, R, DM, DM, alphap, scale, nullptr);

    // q_in / k_in mix in place: bufA = 0.5*ae + 0.5*pe
    mix_kernel<<<(BQ * DM + 255) / 256, blk, 0, stream>>>(bufA, pe_q, BQ * DM);
    mix_kernel<<<(NK * DM + 255) / 256, blk, 0, stream>>>(
        bufA + (size_t)BQ * DM, pe_kv, NK * DM);

    // query/key projection + residual: bufB = bufA @ W + bufA
    gemm_wmma<2><<<dim3(DM / 128, BQ / 64), blk, 0, stream>>>(
        bufA, Wq, nullptr, bufB, BQ, DM, DM, alphap, scale, nullptr);
    gemm_wmma<2><<<dim3(DM / 128, NK / 64), blk, 0, stream>>>(
        bufA + (size_t)BQ * DM, Wk, nullptr, bufB + (size_t)BQ * DM,
        NK, DM, DM, alphap, scale, nullptr);

    // pe_sims = (pe_q @ pe_kv^T) / 16  -> stored in d_out (same shape)
    gemm_wmma<3><<<dim3(NK / 128, BQ / 64), blk, 0, stream>>>(
        pe_q, pe_kv, nullptr, out, BQ, NK, DM, alphap, scale, nullptr);

    // per-row 64th largest of pe_sims
    select_kernel<<<BQ, blk, 0, stream>>>(out, thr, NK, 64);

    // att = mask(pe_sims < thr, 0, (query @ key^T)/16) -> overwrite d_out
    gemm_wmma<4><<<dim3(NK / 128, BQ / 64), blk, 0, stream>>>(
        bufB, bufB + (size_t)BQ * DM, nullptr, out, BQ, NK, DM,
        alphap, scale, thr);
}